// AAO_UNet_45543833206933
// MI455X (gfx1250) — compile-verified
//
#include <hip/hip_runtime.h>
#include <math.h>

// ---------------------------------------------------------------------------
// AAO spectral U-Net forward for MI455X (gfx1250), fp32 end-to-end.
// GEMM-shaped stages (truncated DFT/IDFT, channel MLPs) use
// V_WMMA_F32_16X16X4_F32 with most fragments loaded as single aligned b64
// ops from conflict-free padded LDS / global layouts. The per-mode 160x160
// complex mixing is bandwidth-bound (210 MB of A per forward) and uses
// coalesced b32 streams + VALU fma (N=1 GEMV: matrix cores can't help).
// ---------------------------------------------------------------------------

typedef float v2f __attribute__((ext_vector_type(2)));
typedef float v8f __attribute__((ext_vector_type(8)));

#define LAYERS 4
#define LEVELS 4
#define PRCH 40
#define NP 160
#define MODES 16
#define TWO_PI 6.283185307179586f

__device__ __forceinline__ v8f wmma4(v2f a, v2f b, v8f c) {
#if defined(__gfx1250__) && __has_builtin(__builtin_amdgcn_wmma_f32_16x16x4_f32)
  // D = A(16x4 f32) * B(4x16 f32) + C(16x16 f32)
  return __builtin_amdgcn_wmma_f32_16x16x4_f32(false, a, false, b,
                                               (short)0, c, false, false);
#else
  (void)a; (void)b;
  return c;
#endif
}

__device__ __forceinline__ float gelu_f(float v) {
  float v3 = v * v * v;
  return 0.5f * v * (1.0f + tanhf(0.7978845608028654f * (v + 0.044715f * v3)));
}

// ---------------------------------------------------------------------------
// Encoder: feats = concat(u,x); per level l: f[l] = enc_w[l] @ feats[::2^l] + b
// ---------------------------------------------------------------------------
__global__ void enc_kernel(const float* __restrict__ u, const float* __restrict__ x,
                           const float* __restrict__ enc_w, const float* __restrict__ enc_b,
                           float* __restrict__ f) {
  const long fb[4] = {0, 2621440, 3276800, 3440640};
  int l = blockIdx.y;
  int Hl = 256 >> l;
  int Pl = Hl * Hl;
  long n = (long)PRCH * Pl;
  long idx = (long)blockIdx.x * blockDim.x + threadIdx.x;
  if (idx >= n) return;
  int c = (int)(idx / Pl);
  int p = (int)(idx % Pl);
  int h = p / Hl, w = p % Hl;
  int off = (h << l) * 256 + (w << l);
  float ft[5];
  ft[0] = u[off];
  ft[1] = u[65536 + off];
  ft[2] = u[131072 + off];
  ft[3] = x[off];
  ft[4] = x[65536 + off];
  const float* wv = enc_w + (l * PRCH + c) * 5;
  float acc = enc_b[l * PRCH + c];
#pragma unroll
  for (int i = 0; i < 5; i++) acc = fmaf(wv[i], ft[i], acc);
  f[fb[l] + idx] = acc;
}

// ---------------------------------------------------------------------------
// DFT stage 1: temp[(c,h),ky] = (1/(H*W)) * sum_w f[(c,h),w] * e^{-i 2pi ky w/W}
// GEMM: M=(c,h) rows (16/tile, 8 tiles/WG), K=W, N=16(ky).
// Twiddles transposed [ky][w] with stride 260 (=4*65): b64 frag loads hit all
// 64 LDS banks across the two lane halves -> conflict-free.
// ---------------------------------------------------------------------------
__global__ void dft1_kernel(const float* __restrict__ f, float* __restrict__ t_re,
                            float* __restrict__ t_im, int Hl) {
  __shared__ float twR[16 * 260];
  __shared__ float twI[16 * 260];
  int Wl = Hl;
  int tid = threadIdx.x;
  float scale = 1.0f / (float)(Hl * Wl);
  for (int idx = tid; idx < Wl * MODES; idx += blockDim.x) {
    int w = idx & (Wl - 1);
    int ky = idx / Wl;
    float ang = -TWO_PI * (float)(ky * w) / (float)Wl;
    float s, c;
    sincosf(ang, &s, &c);
    twR[ky * 260 + w] = c * scale;
    twI[ky * 260 + w] = s * scale;
  }
  __syncthreads();
  int lane = tid & 31, wave = tid >> 5;
  int r0 = (blockIdx.x * 8 + wave) * 16;
  int mrow = lane & 15;
  int kp = (lane >> 4) << 1;  // lanes 16-31 carry K+2,K+3
  const float* arow = f + (long)(r0 + mrow) * Wl;
  const float* brow_r = &twR[mrow * 260];
  const float* brow_i = &twI[mrow * 260];
  v8f acc_re = {}, acc_im = {};
  for (int k = 0; k < Wl; k += 4) {
    v2f a = *(const v2f*)(arow + k + kp);
    v2f br = *(const v2f*)(brow_r + k + kp);
    v2f bi = *(const v2f*)(brow_i + k + kp);
    acc_re = wmma4(a, br, acc_re);
    acc_im = wmma4(a, bi, acc_im);
  }
  int mb = (lane >> 4) ? 8 : 0;
#pragma unroll
  for (int j = 0; j < 8; j++) {
    long row = r0 + mb + j;
    t_re[row * 16 + mrow] = acc_re[j];
    t_im[row * 16 + mrow] = acc_im[j];
  }
}

// ---------------------------------------------------------------------------
// DFT stage 2: C[c,r,ky] = sum_h temp[c,h,ky] * e^{-i 2pi (r-8) h/H}
// Per channel a 16 x 16 x H complex GEMM. E stored transposed [r][h] with
// stride Hl+4 (=4*odd) -> aligned conflict-free b64 A-fragment loads.
// ---------------------------------------------------------------------------
__global__ void dft2_kernel(const float* __restrict__ t_re, const float* __restrict__ t_im,
                            float* __restrict__ co_re, float* __restrict__ co_im,
                            int Hl, int qbase) {
  __shared__ float ErT[16 * 260];
  __shared__ float EiT[16 * 260];
  int S = Hl + 4;
  int tid = threadIdx.x;
  for (int idx = tid; idx < Hl * 16; idx += blockDim.x) {
    int h = idx & (Hl - 1);
    int r = idx / Hl;
    float ang = -TWO_PI * (float)((r - 8) * h) / (float)Hl;
    float s, c;
    sincosf(ang, &s, &c);
    ErT[r * S + h] = c;
    EiT[r * S + h] = s;
  }
  __syncthreads();
  int lane = tid & 31, wave = tid >> 5;
  int ch = blockIdx.x * 8 + wave;
  int mrow = lane & 15;
  int kp = (lane >> 4) << 1;
  const float* Tr = t_re + (long)ch * Hl * 16;
  const float* Ti = t_im + (long)ch * Hl * 16;
  const float* arow_r = &ErT[mrow * S];
  const float* arow_i = &EiT[mrow * S];
  v8f acc_re = {}, acc_im = {};
  for (int k = 0; k < Hl; k += 4) {
    v2f ar = *(const v2f*)(arow_r + k + kp);
    v2f ai = *(const v2f*)(arow_i + k + kp);
    v2f br, bi, bin;
    br.x = Tr[(k + kp) * 16 + mrow];
    br.y = Tr[(k + kp + 1) * 16 + mrow];
    bi.x = Ti[(k + kp) * 16 + mrow];
    bi.y = Ti[(k + kp + 1) * 16 + mrow];
    bin.x = -bi.x; bin.y = -bi.y;
    acc_re = wmma4(ar, br, acc_re);
    acc_re = wmma4(ai, bin, acc_re);
    acc_im = wmma4(ar, bi, acc_im);
    acc_im = wmma4(ai, br, acc_im);
  }
  long q = qbase + ch;
  int mb = (lane >> 4) ? 8 : 0;
#pragma unroll
  for (int j = 0; j < 8; j++) {
    int r = mb + j;
    co_re[q * 256 + r * 16 + mrow] = acc_re[j];
    co_im[q * 256 + r * 16 + mrow] = acc_im[j];
  }
}

// ---------------------------------------------------------------------------
// Mode mixing: mixed[p,m] = sum_q A[p,q,m] * coeff[q,m]   (complex, per mode)
// Pure HBM-bandwidth kernel: lanes <-> modes, A streamed fully coalesced.
// ---------------------------------------------------------------------------
__global__ void mix_kernel(const float* __restrict__ A_re, const float* __restrict__ A_im,
                           const float* __restrict__ co_re, const float* __restrict__ co_im,
                           float* __restrict__ mx_re, float* __restrict__ mx_im) {
  int m = threadIdx.x;  // 256 modes
  int p = blockIdx.x;   // 160 output rows
  const float* ar = A_re + (long)p * NP * 256 + m;
  const float* ai = A_im + (long)p * NP * 256 + m;
  float accr = 0.f, acci = 0.f;
  for (int q = 0; q < NP; q++) {
    float av = ar[(long)q * 256];
    float bv = ai[(long)q * 256];
    float cr = co_re[q * 256 + m];
    float ci = co_im[q * 256 + m];
    accr = fmaf(av, cr, fmaf(-bv, ci, accr));
    acci = fmaf(av, ci, fmaf(bv, cr, acci));
  }
  mx_re[p * 256 + m] = accr;
  mx_im[p * 256 + m] = acci;
}

// ---------------------------------------------------------------------------
// IDFT stage 1: s1[c,h,ky] = sum_r mixed[q,r,ky] * e^{+i 2pi (r-8) h/H}
// K=16 contraction; twiddle A-fragments computed in-register via sincosf.
// Output interleaved per row: s_c[row*32 + {0..15:re | 16..31:im}] so stage 2
// reads its A-fragments as single b64 loads without branching.
// ---------------------------------------------------------------------------
__global__ void idft1_kernel(const float* __restrict__ mx_re, const float* __restrict__ mx_im,
                             float* __restrict__ s_c, int Hl, int qbase) {
  int tid = threadIdx.x;
  int lane = tid & 31, wave = tid >> 5;
  int gw = blockIdx.x * 8 + wave;
  int tilesPerCh = Hl >> 4;
  int ch = gw / tilesPerCh;
  int ht = gw % tilesPerCh;
  int h = ht * 16 + (lane & 15);
  int mrow = lane & 15;
  int kp = (lane >> 4) << 1;
  long q = qbase + ch;
  const float* Br = mx_re + q * 256;
  const float* Bi = mx_im + q * 256;
  float base_ang = TWO_PI * (float)h / (float)Hl;
  v8f acc_re = {}, acc_im = {};
  for (int k = 0; k < 16; k += 4) {
    v2f ar, ai, br, bi, bin;
    float s0, c0, s1, c1;
    sincosf(base_ang * (float)(k + kp - 8), &s0, &c0);
    sincosf(base_ang * (float)(k + kp - 7), &s1, &c1);
    ar.x = c0; ar.y = c1;
    ai.x = s0; ai.y = s1;
    br.x = Br[(k + kp) * 16 + mrow];  br.y = Br[(k + kp + 1) * 16 + mrow];
    bi.x = Bi[(k + kp) * 16 + mrow];  bi.y = Bi[(k + kp + 1) * 16 + mrow];
    bin.x = -bi.x; bin.y = -bi.y;
    acc_re = wmma4(ar, br, acc_re);
    acc_re = wmma4(ai, bin, acc_re);
    acc_im = wmma4(ar, bi, acc_im);
    acc_im = wmma4(ai, br, acc_im);
  }
  int mb = (lane >> 4) ? 8 : 0;
  long rowbase = (long)ch * Hl + ht * 16;
#pragma unroll
  for (int j = 0; j < 8; j++) {
    long row = rowbase + mb + j;
    s_c[row * 32 + mrow] = acc_re[j];
    s_c[row * 32 + 16 + mrow] = acc_im[j];
  }
}

// ---------------------------------------------------------------------------
// IDFT stage 2 (hermitian irfft along w):
// g[(c,h),w] = Re(s1[..,0]) + sum_{ky>=1} 2*(s_re*cos - s_im*sin)
// GEMM K=32 (re|im), N=w tiles. A-fragments hoisted across N tiles; B table
// transposed [w][k] stride 36 (=4*9) -> conflict-free b64 loads.
// ---------------------------------------------------------------------------
__global__ void idft2_kernel(const float* __restrict__ s_c, float* __restrict__ g, int Hl) {
  __shared__ float CtT[256 * 36];  // [w][kk]: kk<16 -> Cr, kk>=16 -> Ci
  int Wl = Hl;
  int tid = threadIdx.x;
  for (int idx = tid; idx < 16 * Wl; idx += blockDim.x) {
    int ky = idx / Wl, w = idx & (Wl - 1);
    float ang = TWO_PI * (float)(ky * w) / (float)Wl;
    float s, c;
    sincosf(ang, &s, &c);
    CtT[w * 36 + ky] = (ky == 0) ? 1.f : 2.f * c;
    CtT[w * 36 + 16 + ky] = (ky == 0) ? 0.f : -2.f * s;
  }
  __syncthreads();
  int lane = tid & 31, wave = tid >> 5;
  int r0 = (blockIdx.x * 8 + wave) * 16;
  int mrow = lane & 15;
  int kp = (lane >> 4) << 1;
  int mb = (lane >> 4) ? 8 : 0;
  const float* Arow = s_c + (long)(r0 + mrow) * 32;
  v2f af[8];
#pragma unroll
  for (int t = 0; t < 8; t++) af[t] = *(const v2f*)(Arow + t * 4 + kp);
  for (int n0 = 0; n0 < Wl; n0 += 16) {
    const float* brow = &CtT[(n0 + mrow) * 36];
    v8f acc = {};
#pragma unroll
    for (int t = 0; t < 8; t++) {
      v2f b = *(const v2f*)(brow + t * 4 + kp);
      acc = wmma4(af[t], b, acc);
    }
#pragma unroll
    for (int j = 0; j < 8; j++) {
      long row = r0 + mb + j;
      g[row * (long)Wl + n0 + mrow] = acc[j];
    }
  }
}

// ---------------------------------------------------------------------------
// Channel MLP: f = g + gelu(W2 @ gelu(W1 @ g + b1) + b2), per pixel.
// 128 pixels / WG. Activations stored transposed [pixel][ch] and weights
// [row][ch], both stride 44 (=4*11): every A/B fragment is one aligned,
// bank-conflict-free ds_load_b64. Total LDS ~62.5 KB.
// ---------------------------------------------------------------------------
__global__ void mlp_kernel(const float* __restrict__ g, float* __restrict__ f,
                           const float* __restrict__ w1, const float* __restrict__ b1,
                           const float* __restrict__ w2, const float* __restrict__ b2,
                           int Pl) {
  __shared__ float W1[48 * 44];
  __shared__ float W2[48 * 44];
  __shared__ float B1[48];
  __shared__ float B2[48];
  __shared__ float Ax[128 * 44];    // [pixel][ch]
  __shared__ float Amid[128 * 44];  // [pixel][ch]
  int tid = threadIdx.x;
  for (int idx = tid; idx < 48 * 40; idx += blockDim.x) {
    int r = idx / 40, c = idx - r * 40;
    W1[r * 44 + c] = (r < 40) ? w1[r * 40 + c] : 0.f;
    W2[r * 44 + c] = (r < 40) ? w2[r * 40 + c] : 0.f;
  }
  for (int idx = tid; idx < 48; idx += blockDim.x) {
    B1[idx] = (idx < 40) ? b1[idx] : 0.f;
    B2[idx] = (idx < 40) ? b2[idx] : 0.f;
  }
  long p0 = (long)blockIdx.x * 128;
  for (int idx = tid; idx < 40 * 128; idx += blockDim.x) {
    int ch = idx >> 7, px = idx & 127;
    Ax[px * 44 + ch] = g[(long)ch * Pl + p0 + px];
  }
  __syncthreads();
  int lane = tid & 31, wave = tid >> 5;
  int mrow = lane & 15;
  int kp = (lane >> 4) << 1;
  int n = wave;  // pixel tile 0..7
  int mb = (lane >> 4) ? 8 : 0;
  int px = n * 16 + mrow;
  const float* brow1 = &Ax[px * 44];
  // GEMM1 + gelu -> Amid
  for (int m = 0; m < 3; m++) {
    const float* arow = &W1[(m * 16 + mrow) * 44];
    v8f acc = {};
    for (int k = 0; k < 40; k += 4) {
      v2f a = *(const v2f*)(arow + k + kp);
      v2f b = *(const v2f*)(brow1 + k + kp);
      acc = wmma4(a, b, acc);
    }
#pragma unroll
    for (int j = 0; j < 8; j++) {
      int ch = m * 16 + mb + j;
      if (ch < 40) Amid[px * 44 + ch] = gelu_f(acc[j] + B1[ch]);
    }
  }
  __syncthreads();
  // GEMM2 + gelu + residual -> f
  const float* brow2 = &Amid[px * 44];
  for (int m = 0; m < 3; m++) {
    const float* arow = &W2[(m * 16 + mrow) * 44];
    v8f acc = {};
    for (int k = 0; k < 40; k += 4) {
      v2f a = *(const v2f*)(arow + k + kp);
      v2f b = *(const v2f*)(brow2 + k + kp);
      acc = wmma4(a, b, acc);
    }
#pragma unroll
    for (int j = 0; j < 8; j++) {
      int ch = m * 16 + mb + j;
      if (ch < 40) {
        f[(long)ch * Pl + p0 + px] = Ax[px * 44 + ch] + gelu_f(acc[j] + B2[ch]);
      }
    }
  }
}

// ---------------------------------------------------------------------------
// Decoder: out[l] = dec_w[l] @ f[l] + dec_b[l]  (40 -> 1 channels)
// ---------------------------------------------------------------------------
__global__ void dec_kernel(const float* __restrict__ f, const float* __restrict__ dec_w,
                           const float* __restrict__ dec_b, float* __restrict__ out) {
  const long fb[4] = {0, 2621440, 3276800, 3440640};
  const long ob[4] = {0, 65536, 81920, 86016};
  int l = blockIdx.y;
  int Hl = 256 >> l;
  int Pl = Hl * Hl;
  int p = blockIdx.x * blockDim.x + threadIdx.x;
  if (p >= Pl) return;
  const float* fl = f + fb[l];
  const float* wv = dec_w + l * 40;
  float acc = dec_b[l];
  for (int c = 0; c < 40; c++) acc = fmaf(wv[c], fl[(long)c * Pl + p], acc);
  out[ob[l] + p] = acc;
}

// ---------------------------------------------------------------------------
extern "C" void kernel_launch(void* const* d_in, const int* in_sizes, int n_in,
                              void* d_out, int out_size, void* d_ws, size_t ws_size,
                              hipStream_t stream) {
  const float* u = (const float*)d_in[0];
  const float* x = (const float*)d_in[1];
  const float* enc_w = (const float*)d_in[2];
  const float* enc_b = (const float*)d_in[3];
  const float* dec_w = (const float*)d_in[4];
  const float* dec_b = (const float*)d_in[5];
  const float* c1_w = (const float*)d_in[6];
  const float* c1_b = (const float*)d_in[7];
  const float* c2_w = (const float*)d_in[8];
  const float* c2_b = (const float*)d_in[9];
  const float* A_re = (const float*)d_in[10];
  const float* A_im = (const float*)d_in[11];
  (void)in_sizes; (void)n_in; (void)out_size; (void)ws_size;

  float* ws = (float*)d_ws;
  float* f = ws;                    // 40*87040 = 3,481,600 floats
  float* g = f + 3481600;           // 3,481,600
  float* t_re = g + 3481600;        // 19200 rows * 16
  float* t_im = t_re + 307200;
  float* s_c = t_re;                // idft1 interleaved output reuses t region
  float* co_re = t_im + 307200;     // 160*256
  float* co_im = co_re + 40960;
  float* mx_re = co_im + 40960;
  float* mx_im = mx_re + 40960;     // total 7,741,440 floats (~31 MB)

  const long fb[4] = {0, 2621440, 3276800, 3440640};
  const int troff[4] = {0, 10240, 15360, 17920};  // row offsets into temp/s_c
  const int Hs[4] = {256, 128, 64, 32};

  enc_kernel<<<dim3((40 * 65536 + 255) / 256, 4), 256, 0, stream>>>(u, x, enc_w, enc_b, f);

  for (int i = 0; i < LAYERS; i++) {
    for (int l = 0; l < LEVELS; l++) {
      int Hl = Hs[l];
      dft1_kernel<<<dim3((40 * Hl) / 128), 256, 0, stream>>>(
          f + fb[l], t_re + (long)troff[l] * 16, t_im + (long)troff[l] * 16, Hl);
    }
    for (int l = 0; l < LEVELS; l++) {
      dft2_kernel<<<dim3(5), 256, 0, stream>>>(
          t_re + (long)troff[l] * 16, t_im + (long)troff[l] * 16, co_re, co_im, Hs[l], l * 40);
    }
    mix_kernel<<<dim3(160), 256, 0, stream>>>(
        A_re + (long)i * NP * NP * 256, A_im + (long)i * NP * NP * 256,
        co_re, co_im, mx_re, mx_im);
    for (int l = 0; l < LEVELS; l++) {
      int Hl = Hs[l];
      int nwaves = 40 * (Hl >> 4);
      idft1_kernel<<<dim3(nwaves / 8), 256, 0, stream>>>(
          mx_re, mx_im, s_c + (long)troff[l] * 32, Hl, l * 40);
    }
    for (int l = 0; l < LEVELS; l++) {
      int Hl = Hs[l];
      idft2_kernel<<<dim3((40 * Hl) / 128), 256, 0, stream>>>(
          s_c + (long)troff[l] * 32, g + fb[l], Hl);
    }
    for (int l = 0; l < LEVELS; l++) {
      int Pl = Hs[l] * Hs[l];
      const float* w1 = c1_w + (long)(i * 4 + l) * 40 * 40;
      const float* bb1 = c1_b + (long)(i * 4 + l) * 40;
      const float* w2 = c2_w + (long)(i * 4 + l) * 40 * 40;
      const float* bb2 = c2_b + (long)(i * 4 + l) * 40;
      mlp_kernel<<<dim3(Pl / 128), 256, 0, stream>>>(g + fb[l], f + fb[l], w1, bb1, w2, bb2, Pl);
    }
  }

  dec_kernel<<<dim3((65536 + 255) / 256, 4), 256, 0, stream>>>(f, dec_w, dec_b, (float*)d_out);
}